// QuanvolutionHybridClassifier_65481071396868
// MI455X (gfx1250) — compile-verified
//
#include <hip/hip_runtime.h>
#include <math.h>

typedef float v2f __attribute__((ext_vector_type(2)));
typedef float v8f __attribute__((ext_vector_type(8)));

#define NBATCH  8192
#define NC      10
#define FLATN   784
#define COMBN   1568
#define CSTR    1576      // padded LDS row stride in floats
#define TILE_M  16        // images per workgroup
#define THREADS 128
#define NWAVES  4
#define KSTEPS  392       // 1568 / 4
#define KPW     98        // k-steps per wave

// ---- 4-qubit real statevector helpers (fully unrolled, constant indices) ----
// wire w -> bit (8 >> w):  wire0=8, wire1=4, wire2=2, wire3=1

template<int BIT>
__device__ __forceinline__ void ry_gate(float st[16], float c, float s) {
#pragma unroll
  for (int i = 0; i < 16; ++i) {
    if ((i & BIT) == 0) {
      float a0 = st[i], a1 = st[i | BIT];
      st[i]       = c * a0 - s * a1;
      st[i | BIT] = s * a0 + c * a1;
    }
  }
}

template<int CB, int TB>
__device__ __forceinline__ void cnot_gate(float st[16]) {
#pragma unroll
  for (int i = 0; i < 16; ++i) {
    if ((i & CB) != 0 && (i & TB) == 0) {
      float t = st[i]; st[i] = st[i | TB]; st[i | TB] = t;
    }
  }
}

__global__ __launch_bounds__(THREADS)
void quanv_fused_kernel(const float* __restrict__ x,
                        const float* __restrict__ conv_w,
                        const float* __restrict__ conv_b,
                        const float* __restrict__ qp,
                        const float* __restrict__ lin_w,
                        const float* __restrict__ lin_b,
                        float* __restrict__ out) {
  extern __shared__ float lds[];
  float* comb = lds;                         // [TILE_M][CSTR]
  float* part = lds + TILE_M * CSTR;         // [NWAVES][16][16]

  const int tid  = threadIdx.x;
  const int wave = tid >> 5;
  const int lane = tid & 31;
  const int img0 = blockIdx.x * TILE_M;

  // uniform conv weights -> registers (scalar loads)
  float w[16], cb[4];
#pragma unroll
  for (int i = 0; i < 16; ++i) w[i] = conv_w[i];
#pragma unroll
  for (int i = 0; i < 4; ++i)  cb[i] = conv_b[i];

  // ---------- Phase 1: 2x2/stride-2 conv -> comb[:, 0:784] ----------
  for (int idx = tid; idx < TILE_M * 196; idx += THREADS) {
    int img = idx / 196;
    int pos = idx - img * 196;
    int oy = pos / 14, ox = pos - oy * 14;
    const float* xb = x + (size_t)(img0 + img) * FLATN + (oy * 2) * 28 + ox * 2;
    float p00 = xb[0], p01 = xb[1], p10 = xb[28], p11 = xb[29];
    float* crow = comb + img * CSTR;
#pragma unroll
    for (int ch = 0; ch < 4; ++ch) {
      float v = fmaf(w[ch * 4 + 0], p00,
                fmaf(w[ch * 4 + 1], p01,
                fmaf(w[ch * 4 + 2], p10,
                fmaf(w[ch * 4 + 3], p11, cb[ch]))));
      crow[ch * 196 + pos] = v;
    }
  }

  // RY gate params (uniform); native trans ops
  float qc[5], qs[5];
#pragma unroll
  for (int i = 0; i < 5; ++i) {
    float h = qp[i] * 0.5f;
    qc[i] = __cosf(h); qs[i] = __sinf(h);
  }

  __syncthreads();

  // ---------- Phase 2: quantum features -> comb[:, 784:1568] ----------
  for (int idx = tid; idx < TILE_M * 196; idx += THREADS) {
    int img = idx / 196;
    int p   = idx - img * 196;
    int pi  = p / 14, pj = p - pi * 14;
    float* crow = comb + img * CSTR;
    int f0 = pi * 56 + pj * 2;           // flat viewed as [14,2,14,2]
    float d0 = crow[f0],      d1 = crow[f0 + 1];
    float d2 = crow[f0 + 28], d3 = crow[f0 + 29];

    // hardware v_cos_f32 / v_sin_f32 (TRANS, co-executes with VALU)
    float v0[2] = {__cosf(0.5f * d0), __sinf(0.5f * d0)};
    float v1[2] = {__cosf(0.5f * d1), __sinf(0.5f * d1)};
    float v2[2] = {__cosf(0.5f * d2), __sinf(0.5f * d2)};
    float v3[2] = {__cosf(0.5f * d3), __sinf(0.5f * d3)};

    float st[16];
#pragma unroll
    for (int s = 0; s < 16; ++s)
      st[s] = v0[(s >> 3) & 1] * v1[(s >> 2) & 1] * v2[(s >> 1) & 1] * v3[s & 1];

    ry_gate<8>(st, qc[0], qs[0]);   // RY(p0) wire0
    ry_gate<4>(st, qc[1], qs[1]);   // RY(p1) wire1
    cnot_gate<8, 4>(st);            // CNOT(0,1)
    ry_gate<2>(st, qc[2], qs[2]);   // RY(p2) wire2
    ry_gate<1>(st, qc[3], qs[3]);   // RY(p3) wire3
    cnot_gate<2, 1>(st);            // CNOT(2,3)
    ry_gate<4>(st, qc[4], qs[4]);   // RY(p4) wire1
    cnot_gate<4, 2>(st);            // CNOT(1,2)

    float z0 = 0.f, z1 = 0.f, z2 = 0.f, z3 = 0.f;
#pragma unroll
    for (int s = 0; s < 16; ++s) {
      float pr = st[s] * st[s];
      z0 += (s & 8) ? -pr : pr;
      z1 += (s & 4) ? -pr : pr;
      z2 += (s & 2) ? -pr : pr;
      z3 += (s & 1) ? -pr : pr;
    }
    float* qdst = crow + FLATN + p * 4;
    qdst[0] = z0; qdst[1] = z1; qdst[2] = z2; qdst[3] = z3;
  }

  __syncthreads();

  // ---------- Phase 3: [16 x 1568] @ [1568 x 16(pad of 10)] via WMMA f32 ----
  // A layout (16x4 f32): lane L -> M = L&15, a.x = A[M][2*(L>>4)], a.y = next K
  // B layout (4x16 f32): lane L -> N = L&15, b.x = B[2*(L>>4)][N] = lin_w[N][k]
  v8f acc = {};
  const int  r     = lane & 15;   // M row (A) and N col (B)
  const int  khalf = lane >> 4;   // which K pair this half-wave owns
  const bool valid = (r < NC);
  // Clamp to a real row so loads are unconditional (no EXEC divergence);
  // zero-select afterwards with v_cndmask.
  const float* wrow = lin_w + (size_t)(valid ? r : 0) * COMBN;
  const float* arow = comb + r * CSTR;

  for (int t = wave * KPW; t < (wave + 1) * KPW; ++t) {
    int k = t * 4 + khalf * 2;
    v2f a, b;
    a.x = arow[k];
    a.y = arow[k + 1];
    float bx = wrow[k];
    float by = wrow[k + 1];
    b.x = valid ? bx : 0.0f;
    b.y = valid ? by : 0.0f;
    acc = __builtin_amdgcn_wmma_f32_16x16x4_f32(
        /*neg_a=*/false, a, /*neg_b=*/false, b,
        /*c_mod=*/(short)0, acc, /*reuse_a=*/false, /*reuse_b=*/false);
  }

  // C/D layout: VGPR v at lane L -> M = v + 8*(L>>4), N = L&15
  float* pw = part + wave * 256;
#pragma unroll
  for (int v = 0; v < 8; ++v) {
    int M = v + khalf * 8;
    pw[M * 16 + r] = acc[v];
  }

  __syncthreads();

  // ---------- Phase 4: reduce partials, bias, log-softmax, store ----------
  if (tid < TILE_M) {
    float logits[NC];
#pragma unroll
    for (int n = 0; n < NC; ++n) {
      float s = lin_b[n];
#pragma unroll
      for (int wv = 0; wv < NWAVES; ++wv)
        s += part[wv * 256 + tid * 16 + n];
      logits[n] = s;
    }
    float m = logits[0];
#pragma unroll
    for (int n = 1; n < NC; ++n) m = fmaxf(m, logits[n]);
    float se = 0.f;
#pragma unroll
    for (int n = 0; n < NC; ++n) se += __expf(logits[n] - m);
    float lse = m + __logf(se);
    float* orow = out + (size_t)(img0 + tid) * NC;
#pragma unroll
    for (int n = 0; n < NC; ++n) orow[n] = logits[n] - lse;
  }
}

extern "C" void kernel_launch(void* const* d_in, const int* in_sizes, int n_in,
                              void* d_out, int out_size, void* d_ws, size_t ws_size,
                              hipStream_t stream) {
  (void)in_sizes; (void)n_in; (void)out_size; (void)d_ws; (void)ws_size;
  const float* x      = (const float*)d_in[0];
  const float* conv_w = (const float*)d_in[1];
  const float* conv_b = (const float*)d_in[2];
  const float* qprm   = (const float*)d_in[3];
  const float* lin_w  = (const float*)d_in[4];
  const float* lin_b  = (const float*)d_in[5];
  float* out = (float*)d_out;

  size_t shmem = (size_t)(TILE_M * CSTR + NWAVES * 256) * sizeof(float);
  dim3 grid(NBATCH / TILE_M);   // 512 workgroups, each owns 16 images
  dim3 block(THREADS);          // 4 wave32s
  quanv_fused_kernel<<<grid, block, shmem, stream>>>(
      x, conv_w, conv_b, qprm, lin_w, lin_b, out);
}